// TRXOS_48163763258194
// MI455X (gfx1250) — compile-verified
//
#include <hip/hip_runtime.h>

typedef __attribute__((ext_vector_type(2))) float v2f;
typedef __attribute__((ext_vector_type(8))) float v8f;

#define B_      8
#define NSUP    5
#define NMAP    6      // 5 supports + 1 query per batch
#define SEQ     16
#define IN_DIM  512
#define TSS     3
#define OUT_DIM 1152
#define NT      560    // C(16,3)
#define PE_SCALE 0.1f
#define LN_EPS   1e-5f

__device__ __forceinline__ v8f wmma_f32(v2f a, v2f b, v8f c) {
  // D(16x16,f32) = A(16x4,f32) * B(4x16,f32) + C
  return __builtin_amdgcn_wmma_f32_16x16x4_f32(false, a, false, b, (short)0, c,
                                               false, false);
}

// ---------------------------------------------------------------------------
// Kernel A: per (b,map) project the 16 base rows (+PE) against the 3 blocks of
// k_w and v_w:  Yk[b,m,p,r,:] = (x_r + pe_r) @ k_w[p*512:(p+1)*512, :]
// ---------------------------------------------------------------------------
__global__ void __launch_bounds__(512) proj_kernel(
    const float* __restrict__ sup, const float* __restrict__ qry,
    const float* __restrict__ k_w, const float* __restrict__ v_w,
    float* __restrict__ Yk, float* __restrict__ Yv)
{
  __shared__ float ldsA[SEQ * IN_DIM];
  const int bm = blockIdx.x;                 // b*NMAP + map
  const int b = bm / NMAP, mp = bm % NMAP;
  const int tid = threadIdx.x;

  for (int idx = tid; idx < SEQ * IN_DIM; idx += 512) {
    int r = idx >> 9, d = idx & (IN_DIM - 1);
    float x = (mp < NSUP)
      ? sup[(((size_t)b * NSUP + mp) * SEQ + r) * IN_DIM + d]
      : qry[((size_t)b * SEQ + r) * IN_DIM + d];
    int de = d & ~1;                          // even pair base
    float ang = (float)r * __expf(-logf(10000.0f) * (float)de / (float)IN_DIM);
    float pe = ((d & 1) ? __cosf(ang) : __sinf(ang)) * PE_SCALE;
    ldsA[idx] = x + pe;
  }
  __syncthreads();

  const int wave  = tid >> 5, lane = tid & 31;
  const int mrow  = lane & 15;
  const int khalf = (lane >> 4) << 1;

  // 432 tile jobs: {k,v} x 3 parts x (1152/16) n-tiles ; 16 waves round-robin
  const int NJOBS = 2 * TSS * (OUT_DIM / 16);
  for (int job = wave; job < NJOBS; job += 16) {
    const int kv = job / (TSS * (OUT_DIM / 16));
    int rem = job % (TSS * (OUT_DIM / 16));
    const int p  = rem / (OUT_DIM / 16);
    const int n0 = (rem % (OUT_DIM / 16)) * 16;
    const float* W = (kv ? v_w : k_w) + (size_t)p * IN_DIM * OUT_DIM;

    v8f acc = {};
    for (int k0 = 0; k0 < IN_DIM; k0 += 4) {
      v2f a = *(const v2f*)&ldsA[mrow * IN_DIM + k0 + khalf];
      const float* wp = W + (size_t)(k0 + khalf) * OUT_DIM + n0 + mrow;
      v2f bb; bb.x = wp[0]; bb.y = wp[OUT_DIM];
      acc = wmma_f32(a, bb, acc);
    }
    float* dst = (kv ? Yv : Yk) + (((size_t)bm * TSS + p) * SEQ) * OUT_DIM + n0;
    #pragma unroll
    for (int r = 0; r < 8; ++r) {
      int m = r + ((lane >> 4) << 3);
      dst[(size_t)m * OUT_DIM + mrow] = acc[r];
    }
  }
}

// ---------------------------------------------------------------------------
// Kernel B: per (b,map,t) combine the tuple rows, add bias, LayerNorm the K
// path, write K/V feature vectors for supports (sk/sv) and queries (qk/qv).
// ---------------------------------------------------------------------------
__global__ void __launch_bounds__(128) combine_ln_kernel(
    const float* __restrict__ Yk, const float* __restrict__ Yv,
    const float* __restrict__ k_b, const float* __restrict__ v_b,
    const float* __restrict__ gamma, const float* __restrict__ beta,
    float* __restrict__ sk, float* __restrict__ sv,
    float* __restrict__ qk, float* __restrict__ qv)
{
  const int t  = blockIdx.x % NT;
  const int bm = blockIdx.x / NT;
  const int mp = bm % NMAP, b = bm / NMAP;
  const int tid = threadIdx.x;

  // decode lexicographic combination index t -> (i < j < kq)
  int i = 0, rem = t;
  for (; i < SEQ; ++i) { int c = (15 - i) * (14 - i) / 2; if (rem < c) break; rem -= c; }
  int j = i + 1;
  for (; j < SEQ; ++j) { int c = 15 - j; if (rem < c) break; rem -= c; }
  int kq = j + 1 + rem;

  const size_t base = (size_t)bm * TSS * SEQ * OUT_DIM;
  const float* yk0 = Yk + base + ((size_t)0 * SEQ + i ) * OUT_DIM;
  const float* yk1 = Yk + base + ((size_t)1 * SEQ + j ) * OUT_DIM;
  const float* yk2 = Yk + base + ((size_t)2 * SEQ + kq) * OUT_DIM;
  const float* yv0 = Yv + base + ((size_t)0 * SEQ + i ) * OUT_DIM;
  const float* yv1 = Yv + base + ((size_t)1 * SEQ + j ) * OUT_DIM;
  const float* yv2 = Yv + base + ((size_t)2 * SEQ + kq) * OUT_DIM;

  float* voutp = (mp < NSUP)
      ? sv + (((size_t)b * NSUP + mp) * NT + t) * OUT_DIM
      : qv + ((size_t)b * NT + t) * OUT_DIM;
  float* koutp = (mp < NSUP)
      ? sk + (((size_t)b * NSUP + mp) * NT + t) * OUT_DIM
      : qk + ((size_t)b * NT + t) * OUT_DIM;

  float kvals[9];
  float s1 = 0.f, s2 = 0.f;
  #pragma unroll
  for (int q = 0; q < 9; ++q) {
    int d = tid + q * 128;
    float kk = yk0[d] + yk1[d] + yk2[d] + k_b[d];
    kvals[q] = kk; s1 += kk; s2 += kk * kk;
    voutp[d] = yv0[d] + yv1[d] + yv2[d] + v_b[d];
  }

  __shared__ float r1[128], r2[128];
  r1[tid] = s1; r2[tid] = s2;
  __syncthreads();
  for (int s = 64; s > 0; s >>= 1) {
    if (tid < s) { r1[tid] += r1[tid + s]; r2[tid] += r2[tid + s]; }
    __syncthreads();
  }
  float mu   = r1[0] * (1.0f / OUT_DIM);
  float var  = r2[0] * (1.0f / OUT_DIM) - mu * mu;
  float rstd = rsqrtf(var + LN_EPS);
  #pragma unroll
  for (int q = 0; q < 9; ++q) {
    int d = tid + q * 128;
    koutp[d] = (kvals[q] - mu) * rstd * gamma[d] + beta[d];
  }
}

// ---------------------------------------------------------------------------
// Kernel C: scores[b,c,t,s] = <q_ks[b,t,:], cls_k[b,c,s,:]> / sqrt(1152)
// One wave per 16x16 output tile; K = 1152 in steps of 4 (f32 WMMA).
// ---------------------------------------------------------------------------
__global__ void __launch_bounds__(32) scores_kernel(
    const float* __restrict__ qk, const float* __restrict__ sk,
    const int* __restrict__ labels, float* __restrict__ scores)
{
  const int TILES = (NT / 16) * (NT / 16);   // 1225
  int tile = blockIdx.x % TILES;
  int bc   = blockIdx.x / TILES;
  int c = bc % NSUP, b = bc / NSUP;
  int tt = (tile / (NT / 16)) * 16;
  int ss = (tile % (NT / 16)) * 16;
  int cls = labels[c];                        // idx = support_labels[0]

  const int lane  = threadIdx.x;
  const int mrow  = lane & 15;
  const int khalf = (lane >> 4) << 1;

  const float* A  = qk + ((size_t)b * NT + tt + mrow) * OUT_DIM + khalf;
  const float* Bm = sk + (((size_t)b * NSUP + cls) * NT + ss + mrow) * OUT_DIM + khalf;
  v8f acc = {};
  for (int k0 = 0; k0 < OUT_DIM; k0 += 4) {
    v2f a  = *(const v2f*)(A  + k0);
    v2f bb = *(const v2f*)(Bm + k0);
    acc = wmma_f32(a, bb, acc);
  }
  const float scale = 0.029462782549439483f;  // 1/sqrt(1152)
  float* dst = scores + ((size_t)bc * NT + tt) * NT + ss;
  #pragma unroll
  for (int r = 0; r < 8; ++r) {
    int m = r + ((lane >> 4) << 3);
    dst[(size_t)m * NT + mrow] = acc[r] * scale;
  }
}

// ---------------------------------------------------------------------------
// Kernel D: softmax over the t axis (axis=-2): per (b,c), per column s.
// ---------------------------------------------------------------------------
__global__ void __launch_bounds__(256) softmax_kernel(float* __restrict__ scores)
{
  float* S = scores + (size_t)blockIdx.x * NT * NT;
  for (int s = threadIdx.x; s < NT; s += 256) {
    float mx = -3.4e38f;
    for (int t = 0; t < NT; ++t) mx = fmaxf(mx, S[(size_t)t * NT + s]);
    float sum = 0.f;
    for (int t = 0; t < NT; ++t) {
      float e = __expf(S[(size_t)t * NT + s] - mx);
      S[(size_t)t * NT + s] = e;
      sum += e;
    }
    float inv = 1.0f / sum;
    for (int t = 0; t < NT; ++t) S[(size_t)t * NT + s] *= inv;
  }
}

// ---------------------------------------------------------------------------
// Kernel E: P = scores @ cls_v ; diff = q_vs - P ; logits -= sum(diff^2)/560.
// One wave per 16x16 tile of P; fused diff store + wave-reduced atomic logits.
// ---------------------------------------------------------------------------
__global__ void __launch_bounds__(32) proto_diff_kernel(
    const float* __restrict__ scores, const float* __restrict__ sv,
    const float* __restrict__ qv, const int* __restrict__ labels,
    float* __restrict__ logits, float* __restrict__ diffs)
{
  const int TILES = (NT / 16) * (OUT_DIM / 16); // 35*72 = 2520
  int tile = blockIdx.x % TILES;
  int bc   = blockIdx.x / TILES;
  int c = bc % NSUP, b = bc / NSUP;
  int tt = (tile / (OUT_DIM / 16)) * 16;
  int dd = (tile % (OUT_DIM / 16)) * 16;
  int cls = labels[c];

  const int lane  = threadIdx.x;
  const int mrow  = lane & 15;
  const int khalf = (lane >> 4) << 1;

  const float* A  = scores + ((size_t)bc * NT + tt + mrow) * NT + khalf;
  const float* Bv = sv + ((size_t)b * NSUP + cls) * NT * OUT_DIM + dd + mrow;
  v8f acc = {};
  for (int k0 = 0; k0 < NT; k0 += 4) {
    v2f a = *(const v2f*)(A + k0);
    const float* bp = Bv + (size_t)(k0 + khalf) * OUT_DIM;
    v2f bb; bb.x = bp[0]; bb.y = bp[OUT_DIM];
    acc = wmma_f32(a, bb, acc);
  }
  float part = 0.f;
  #pragma unroll
  for (int r = 0; r < 8; ++r) {
    int m = r + ((lane >> 4) << 3);
    int t = tt + m, d = dd + mrow;
    float q  = qv[((size_t)b * NT + t) * OUT_DIM + d];
    float df = q - acc[r];
    diffs[((size_t)bc * NT + t) * OUT_DIM + d] = df;
    part += df * df;
  }
  for (int o = 16; o > 0; o >>= 1) part += __shfl_down(part, o, 32);
  if (lane == 0) atomicAdd(&logits[bc], -part * (1.0f / NT));
}

__global__ void init_logits_kernel(float* __restrict__ logits) {
  if (threadIdx.x < B_ * NSUP) logits[threadIdx.x] = 0.f;
}

// ---------------------------------------------------------------------------
extern "C" void kernel_launch(void* const* d_in, const int* in_sizes, int n_in,
                              void* d_out, int out_size, void* d_ws, size_t ws_size,
                              hipStream_t stream)
{
  const float* sup   = (const float*)d_in[0];
  const float* qry   = (const float*)d_in[1];
  const int*   lbl   = (const int*)d_in[2];   // support_labels; row 0 used
  const float* k_w   = (const float*)d_in[3];
  const float* k_b   = (const float*)d_in[4];
  const float* v_w   = (const float*)d_in[5];
  const float* v_b   = (const float*)d_in[6];
  const float* gamma = (const float*)d_in[7];
  const float* beta  = (const float*)d_in[8];

  float* out    = (float*)d_out;
  float* logits = out;                    // (8,5)
  float* diffs  = out + B_ * NSUP;        // (8,5,560,1152)

  float* ws = (float*)d_ws;
  size_t o = 0;
  float* Yk = ws + o; o += (size_t)B_ * NMAP * TSS * SEQ * OUT_DIM;
  float* Yv = ws + o; o += (size_t)B_ * NMAP * TSS * SEQ * OUT_DIM;
  float* sk = ws + o; o += (size_t)B_ * NSUP * NT * OUT_DIM;
  float* sv = ws + o; o += (size_t)B_ * NSUP * NT * OUT_DIM;
  float* qk = ws + o; o += (size_t)B_ * NT * OUT_DIM;
  float* qv = ws + o; o += (size_t)B_ * NT * OUT_DIM;
  float* sc = ws + o; o += (size_t)B_ * NSUP * NT * NT;
  (void)ws_size; (void)in_sizes; (void)n_in; (void)out_size;

  hipLaunchKernelGGL(init_logits_kernel, dim3(1), dim3(64), 0, stream, logits);
  hipLaunchKernelGGL(proj_kernel, dim3(B_ * NMAP), dim3(512), 0, stream,
                     sup, qry, k_w, v_w, Yk, Yv);
  hipLaunchKernelGGL(combine_ln_kernel, dim3(B_ * NMAP * NT), dim3(128), 0, stream,
                     Yk, Yv, k_b, v_b, gamma, beta, sk, sv, qk, qv);
  hipLaunchKernelGGL(scores_kernel, dim3(B_ * NSUP * 1225), dim3(32), 0, stream,
                     qk, sk, lbl, sc);
  hipLaunchKernelGGL(softmax_kernel, dim3(B_ * NSUP), dim3(256), 0, stream, sc);
  hipLaunchKernelGGL(proto_diff_kernel, dim3(B_ * NSUP * 2520), dim3(32), 0, stream,
                     sc, sv, qv, lbl, logits, diffs);
}